// RoiPoolingConv_523986010557
// MI455X (gfx1250) — compile-verified
//
#include <hip/hip_runtime.h>

// ROI pooling (bilinear crop-resize), MI455X / gfx1250.
// feature_map: [1, 64, 64, 1024] f32   rois: [1, 300, 4] i32 (x,y,w,h)
// out:         [1, 300, 14, 14, 1024] f32
//
// Memory-bound kernel: ~257 MB HBM traffic (16 MB fm read, 241 MB out write)
// => ~11us floor @ 23.3 TB/s. Strategy: B128 coalesced corner loads (fm stays
// resident in 192MB L2), non-temporal B128 stores for the streamed output,
// global_prefetch for the next px column.

typedef float v4f __attribute__((ext_vector_type(4)));

#define POOL 14
#define FM_H 64
#define FM_W 64
#define FM_C 1024
#define NROI 300

__global__ __launch_bounds__(256) void roi_pool_bilinear_kernel(
    const float* __restrict__ fm,    // [FM_H, FM_W, FM_C]
    const int* __restrict__ rois,    // [NROI, 4] = (x, y, w, h)
    float* __restrict__ out)         // [NROI, POOL, POOL, FM_C]
{
    const int blk = blockIdx.x;            // roi * POOL + py
    const int roi = blk / POOL;
    const int py  = blk - roi * POOL;
    const int c   = threadIdx.x * 4;       // 256 threads * 4ch = 1024 channels

    // ROI params are uniform across the block (4 ints, L2-hot after first use).
    const int rx = rois[roi * 4 + 0];
    const int ry = rois[roi * 4 + 1];
    const int rw = rois[roi * 4 + 2];
    const int rh = rois[roi * 4 + 3];

    // Legacy-TF resize: src = dst * (in/out), upper index clamped to edge-1.
    const float sy = (float)rh * (1.0f / (float)POOL);
    const float sx = (float)rw * (1.0f / (float)POOL);

    const float fy = (float)ry + (float)py * sy;
    const int y0   = (int)floorf(fy);
    const int y1   = min(y0 + 1, ry + rh - 1);
    const float wy = fy - (float)y0;

    const int xhi  = rx + rw - 1;          // clamp limit for x1

    const float* __restrict__ row0 = fm + (size_t)y0 * (FM_W * FM_C) + c;
    const float* __restrict__ row1 = fm + (size_t)y1 * (FM_W * FM_C) + c;
    float* __restrict__ orow = out + (size_t)blk * (POOL * FM_C) + c;

    // First column indices.
    int x0 = (int)floorf((float)rx);       // == rx (integer ROI corner)
    int x1 = min(x0 + 1, xhi);
    float wx = 0.0f;                       // fx(0) = rx exactly

    #pragma unroll 2
    for (int px = 0; px < POOL; ++px) {
        // Prefetch next column's corner rows into near caches (gfx1250
        // global_prefetch_b8); fm is L2-resident, this hides L2->L0 latency.
        int nx0 = x0, nx1 = x1;
        float nwx = wx;
        if (px + 1 < POOL) {
            const float fxn = (float)rx + (float)(px + 1) * sx;
            nx0 = (int)floorf(fxn);
            nx1 = min(nx0 + 1, xhi);
            nwx = fxn - (float)nx0;
            __builtin_prefetch(row0 + (size_t)nx0 * FM_C, 0, 3);
            __builtin_prefetch(row1 + (size_t)nx0 * FM_C, 0, 3);
        }

        const v4f tl = *(const v4f*)(row0 + (size_t)x0 * FM_C);
        const v4f tr = *(const v4f*)(row0 + (size_t)x1 * FM_C);
        const v4f bl = *(const v4f*)(row1 + (size_t)x0 * FM_C);
        const v4f br = *(const v4f*)(row1 + (size_t)x1 * FM_C);

        const v4f top = tl + wx * (tr - tl);
        const v4f bot = bl + wx * (br - bl);
        const v4f res = top + wy * (bot - top);

        // Streamed 241MB output: NT store so it doesn't evict the fm from L2.
        __builtin_nontemporal_store(res, (v4f*)(orow + (size_t)px * FM_C));

        x0 = nx0; x1 = nx1; wx = nwx;
    }
}

extern "C" void kernel_launch(void* const* d_in, const int* in_sizes, int n_in,
                              void* d_out, int out_size, void* d_ws, size_t ws_size,
                              hipStream_t stream) {
    const float* fm  = (const float*)d_in[0];   // [1,64,64,1024] f32
    const int* rois  = (const int*)d_in[1];     // [1,300,4] i32
    // d_in[2] = pool_size scalar (14), compile-time constant here.
    float* out = (float*)d_out;                 // [1,300,14,14,1024] f32

    const int grid = NROI * POOL;               // 4200 blocks, 8 waves each
    roi_pool_bilinear_kernel<<<grid, 256, 0, stream>>>(fm, rois, out);
}